// SpecTransform_45337674777250
// MI455X (gfx1250) — compile-verified
//
#include <hip/hip_runtime.h>

// ---------------------------------------------------------------------------
// Blocked parallel biquad chain for MI455X (gfx1250).
//
// y[t] = x[t] + b1*x[t-1] + b2*x[t-2] - a1*y[t-1] - a2*y[t-2], clip([-1,1])
// applied twice per signal. Linear-state decomposition:
//   T = L*C chunks; per chunk compute particular y_p (zero initial y-state,
//   true x history read from the input buffer), then
//   s_c = M^L s_{c-1} + p_c  (2x2, serial over C chunks, staged in LDS),
//   y[cL+i] = y_p[cL+i] + h1[i]*s_prev.u + h2[i]*s_prev.v  -> clip.
// The rank-2 correction is computed with V_WMMA_F32_16X16X4_F32 on
// 16-chunk x 16-sample tiles (K=0: h1/state-u, K=1: h2/state-v, K=2,3 = 0).
// ---------------------------------------------------------------------------

#define NSEQ 128          // 64 x-sequences + 64 n-sequences
#define T_LEN 160000
#define L_CHUNK 400       // chunk length (multiple of 16)
#define C_CHUNK 400       // chunks per sequence (multiple of 16)
#define GROUPS (C_CHUNK / 16)   // 25 chunk-groups of 16 per sequence

typedef float v2f __attribute__((ext_vector_type(2)));
typedef float v8f __attribute__((ext_vector_type(8)));

// ---------------------------------------------------------------------------
// Homogeneous impulse responses h1 (y-1=1,y-2=0) and h2 (y-1=0,y-2=1) for the
// three filter configs: f=0 fixed HP, f=1 rand-x, f=2 rand-n.
// M^L is recovered from h*(L-1), h*(L-2) by the scan kernel.
// ---------------------------------------------------------------------------
__global__ void hp_precompute_h(const float* __restrict__ coefX,
                                const float* __restrict__ coefN,
                                float* __restrict__ hArr) {
    int f = threadIdx.x;
    if (f >= 3) return;
    float a1, a2;
    if (f == 0) { a1 = -1.99599f; a2 = 0.996f; }
    else {
        const float* c = (f == 1) ? coefX : coefN;
        a1 = c[0]; a2 = c[1];
    }
    float* h1 = hArr + (size_t)(f * 2 + 0) * L_CHUNK;
    float* h2 = hArr + (size_t)(f * 2 + 1) * L_CHUNK;
    float u1 = 1.f, u2 = 0.f;   // h1 initial state
    float w1 = 0.f, w2 = 1.f;   // h2 initial state
    for (int i = 0; i < L_CHUNK; ++i) {
        float hu = -a1 * u1 - a2 * u2;
        float hw = -a1 * w1 - a2 * w2;
        h1[i] = hu; h2[i] = hw;
        u2 = u1; u1 = hu;
        w2 = w1; w1 = hw;
    }
}

// ---------------------------------------------------------------------------
// Particular solution per chunk (zero initial y-state, true x history).
// One thread per (seq, chunk): 128*400 = 51200 independent serial walks of
// 400 steps — chain is 2 dependent FMAs/sample, fully latency-overlapped.
// mode 0: HP stage, inputs in0=x rows 0..63 / in1=n rows 0..63, fixed coefs.
// mode 1: rand stage, input in0 = clipped stage-1 buffer, per-seq coefs.
// ---------------------------------------------------------------------------
__global__ void hp_particular(const float* __restrict__ in0,
                              const float* __restrict__ in1,
                              float* __restrict__ out,
                              const float* __restrict__ coefX,
                              const float* __restrict__ coefN,
                              int mode) {
    int g = blockIdx.x * blockDim.x + threadIdx.x;
    if (g >= NSEQ * C_CHUNK) return;
    int seq = g / C_CHUNK;
    int c   = g % C_CHUNK;

    const float* row;
    float a1, a2, b1, b2;
    if (mode == 0) {
        row = (seq < 64) ? (in0 + (size_t)seq * T_LEN)
                         : (in1 + (size_t)(seq - 64) * T_LEN);
        a1 = -1.99599f; a2 = 0.996f; b1 = -2.0f; b2 = 1.0f;
    } else {
        row = in0 + (size_t)seq * T_LEN;
        const float* cf = (seq < 64) ? coefX : coefN;
        a1 = cf[0]; a2 = cf[1]; b1 = cf[2]; b2 = cf[3];
    }

    int t0 = c * L_CHUNK;
    float x1 = (c > 0) ? row[t0 - 1] : 0.f;
    float x2 = (c > 0) ? row[t0 - 2] : 0.f;
    float y1 = 0.f, y2 = 0.f;
    const float* irow = row + t0;
    float*       orow = out + (size_t)seq * T_LEN + t0;

    for (int i = 0; i < L_CHUNK; ++i) {
        if ((i & 15) == 0 && (t0 + i + 64) < T_LEN)
            __builtin_prefetch(irow + i + 64, 0, 1);   // global_prefetch_b8
        float xt = irow[i];
        float y = fmaf(b1, x1,
                  fmaf(b2, x2,
                  fmaf(-a1, y1,
                  fmaf(-a2, y2, xt))));
        orow[i] = y;
        x2 = x1; x1 = xt;
        y2 = y1; y1 = y;
    }
}

// ---------------------------------------------------------------------------
// Serial 2x2 state scan over chunks. One block handles 8 sequences: chunk-end
// particular values are cooperatively gathered into LDS (parallel,
// independent loads), 8 lanes run the 400-step dependent scan purely out of
// LDS, incoming states are stored back coalesced.
// states[(seq*C + c)*2 + {0,1}] = unclipped y at (cL-1, cL-2).
// ---------------------------------------------------------------------------
__global__ void hp_scan(const float* __restrict__ yp,
                        float* __restrict__ states,
                        const float* __restrict__ hArr,
                        int mode) {
    __shared__ float ps[8 * C_CHUNK * 2];
    __shared__ float ss[8 * C_CHUNK * 2];
    int sbase = blockIdx.x * 8;

    for (int t = threadIdx.x; t < 8 * C_CHUNK; t += blockDim.x) {
        int s = t / C_CHUNK, c = t % C_CHUNK;
        const float* row = yp + (size_t)(sbase + s) * T_LEN + (size_t)c * L_CHUNK;
        ps[t * 2 + 0] = row[L_CHUNK - 1];
        ps[t * 2 + 1] = row[L_CHUNK - 2];
    }
    __syncthreads();

    if (threadIdx.x < 8) {
        int s = threadIdx.x;
        int seq = sbase + s;
        int f = (mode == 0) ? 0 : ((seq < 64) ? 1 : 2);
        const float* h1 = hArr + (size_t)(f * 2 + 0) * L_CHUNK;
        const float* h2 = hArr + (size_t)(f * 2 + 1) * L_CHUNK;
        float m00 = h1[L_CHUNK - 1], m01 = h2[L_CHUNK - 1];
        float m10 = h1[L_CHUNK - 2], m11 = h2[L_CHUNK - 2];
        float u = 0.f, v = 0.f;
        for (int c = 0; c < C_CHUNK; ++c) {
            int t = s * C_CHUNK + c;
            ss[t * 2 + 0] = u;
            ss[t * 2 + 1] = v;
            float p0 = ps[t * 2 + 0];
            float p1 = ps[t * 2 + 1];
            float nu = fmaf(m00, u, fmaf(m01, v, p0));
            float nv = fmaf(m10, u, fmaf(m11, v, p1));
            u = nu; v = nv;
        }
    }
    __syncthreads();

    for (int t = threadIdx.x; t < 8 * C_CHUNK * 2; t += blockDim.x)
        states[(size_t)sbase * C_CHUNK * 2 + t] = ss[t];
}

// ---------------------------------------------------------------------------
// Rank-2 correction + clip via V_WMMA_F32_16X16X4_F32, in place.
// One wave per (seq, group of 16 chunks).
//   A (16x4, M=chunk): lanes 0-15 hold {K0=state.u, K1=state.v}; lanes 16-31
//     (K=2,3) are zeroed.  B (4x16, N=time): lanes 0-15 hold {K0=h1, K1=h2};
//     lanes 16-31 (K=2,3) zeroed.  C/D (16x16): vgpr r, lane l ->
//     chunk = r + 8*(l>=16), time = tb + (l&15)  -> two 64B segments/vgpr.
// ---------------------------------------------------------------------------
__global__ void hp_correct(float* __restrict__ buf,
                           const float* __restrict__ states,
                           const float* __restrict__ hArr,
                           int mode) {
    int lane = threadIdx.x & 31;
    int wid  = (blockIdx.x * blockDim.x + threadIdx.x) >> 5;
    if (wid >= NSEQ * GROUPS) return;   // grid sized exactly; whole-wave uniform
    int seq = wid / GROUPS;
    int g   = wid % GROUPS;
    int c0  = g * 16;

    int f = (mode == 0) ? 0 : ((seq < 64) ? 1 : 2);
    const float* h1 = hArr + (size_t)(f * 2 + 0) * L_CHUNK;
    const float* h2 = hArr + (size_t)(f * 2 + 1) * L_CHUNK;

    v2f a;
    if (lane < 16) {
        const float* sp = states + ((size_t)seq * C_CHUNK + c0 + lane) * 2;
        a.x = sp[0]; a.y = sp[1];
    } else {
        a.x = 0.f; a.y = 0.f;           // K=2,3 lanes contribute zero
    }

    float* base = buf + (size_t)seq * T_LEN + (size_t)c0 * L_CHUNK;
    int col  = lane & 15;
    int half = lane >> 4;

    for (int tb = 0; tb < L_CHUNK; tb += 16) {
        v2f b;
        if (lane < 16) { b.x = h1[tb + lane]; b.y = h2[tb + lane]; }
        else           { b.x = 0.f;           b.y = 0.f; }

        v8f cacc;
#pragma unroll
        for (int r = 0; r < 8; ++r)
            cacc[r] = base[(size_t)(r + half * 8) * L_CHUNK + tb + col];

        v8f d = __builtin_amdgcn_wmma_f32_16x16x4_f32(
            /*neg_a=*/false, a, /*neg_b=*/false, b,
            /*c_mod=*/(short)0, cacc, /*reuse_a=*/false, /*reuse_b=*/false);

#pragma unroll
        for (int r = 0; r < 8; ++r) {
            float yv = fminf(fmaxf(d[r], -1.f), 1.f);
            base[(size_t)(r + half * 8) * L_CHUNK + tb + col] = yv;
        }
    }
}

// ---------------------------------------------------------------------------
extern "C" void kernel_launch(void* const* d_in, const int* in_sizes, int n_in,
                              void* d_out, int out_size, void* d_ws, size_t ws_size,
                              hipStream_t stream) {
    const float* x     = (const float*)d_in[0];   // (64, 160000)
    const float* n     = (const float*)d_in[1];   // (64, 160000)
    const float* coefX = (const float*)d_in[2];   // [a1, a2, b1, b2]
    const float* coefN = (const float*)d_in[3];
    float* out = (float*)d_out;                   // (2*64, 160000) flat

    // workspace: stage-1 buffer | h-responses | incoming chunk states
    float* ws_big = (float*)d_ws;                                  // NSEQ*T
    float* ws_h   = ws_big + (size_t)NSEQ * T_LEN;                 // 3*2*L
    float* ws_st  = ws_h + (size_t)3 * 2 * L_CHUNK;                // NSEQ*C*2

    const int PART_THREADS = NSEQ * C_CHUNK;                       // 51200
    const int PART_BLOCKS  = (PART_THREADS + 255) / 256;
    const int CORR_BLOCKS  = (NSEQ * GROUPS) / 8;                  // 400 (8 waves/blk)

    hp_precompute_h<<<1, 64, 0, stream>>>(coefX, coefN, ws_h);

    // ---- Stage 1: fixed high-pass on all 128 sequences -> ws_big ----
    hp_particular<<<PART_BLOCKS, 256, 0, stream>>>(x, n, ws_big, coefX, coefN, 0);
    hp_scan<<<NSEQ / 8, 256, 0, stream>>>(ws_big, ws_st, ws_h, 0);
    hp_correct<<<CORR_BLOCKS, 256, 0, stream>>>(ws_big, ws_st, ws_h, 0);

    // ---- Stage 2: per-signal random biquad -> d_out (x chain | n chain) ----
    hp_particular<<<PART_BLOCKS, 256, 0, stream>>>(ws_big, nullptr, out, coefX, coefN, 1);
    hp_scan<<<NSEQ / 8, 256, 0, stream>>>(out, ws_st, ws_h, 1);
    hp_correct<<<CORR_BLOCKS, 256, 0, stream>>>(out, ws_st, ws_h, 1);
}